// DFConv_87368224735577
// MI455X (gfx1250) — compile-verified
//
#include <hip/hip_runtime.h>
#include <hip/hip_bf16.h>

typedef __attribute__((ext_vector_type(2))) float v2f;
typedef __attribute__((ext_vector_type(8))) float v8f;

#define KDIM 64
#define PP   16384   // H*W
#define HH   128
#define WW   128
#define LDW  68      // LDS row stride (words): conflict-free for b64 reads

// ---------------------------------------------------------------------------
// GEMM: out[M x NCOLS] = A[M x 64] * Wt[NCOLS x 64]^T + bias
// Weights staged in LDS (zero-padded to 64 rows). One wave32 per 16-row
// M-tile computes ALL 4 N-tiles -> 4 independent WMMA chains, 64 WMMAs/wave.
// ---------------------------------------------------------------------------
template <int NCOLS>
__global__ __launch_bounds__(256) void gemm_nt_bias(
    const float* __restrict__ A, const float* __restrict__ Wt,
    const float* __restrict__ bias, float* __restrict__ out, int M)
{
  constexpr int NT = (NCOLS + 15) / 16;   // 4 for 64 and 56
  __shared__ float wlds[64 * LDW];

  int tid = threadIdx.x;
  // Cooperative load of Wt (NCOLS x 64) -> LDS, zero-pad rows >= NCOLS.
  {
    int row  = tid >> 2;          // 0..63
    int col0 = (tid & 3) << 4;    // 0,16,32,48
#pragma unroll
    for (int j = 0; j < 16; j += 4) {
      float4 v = make_float4(0.f, 0.f, 0.f, 0.f);
      if (row < NCOLS) v = *(const float4*)(Wt + row * KDIM + col0 + j);
      *(float4*)(&wlds[row * LDW + col0 + j]) = v;
    }
  }
  __syncthreads();

  int lane   = tid & 31;
  int wv     = tid >> 5;
  int m_tile = blockIdx.x * 8 + wv;
  if (m_tile * 16 >= M) return;

  int nloc  = lane & 15;          // N within tile (also A row within tile)
  int khalf = (lane >> 4) << 1;   // K sub-offset: 0 (lanes 0-15) / 2 (16-31)
  const float* arow = A + (size_t)((m_tile << 4) + nloc) * KDIM;

  v8f acc[NT];
#pragma unroll
  for (int t = 0; t < NT; ++t) {
    int n = t * 16 + nloc;
    float bv = (n < NCOLS) ? bias[n] : 0.0f;
    v8f ini = {bv, bv, bv, bv, bv, bv, bv, bv};
    acc[t] = ini;
  }

#pragma unroll
  for (int k = 0; k < KDIM; k += 4) {
    v2f a = *(const v2f*)(arow + k + khalf);
#pragma unroll
    for (int t = 0; t < NT; ++t) {
      v2f b = *(const v2f*)(&wlds[(t * 16 + nloc) * LDW + k + khalf]);
      acc[t] = __builtin_amdgcn_wmma_f32_16x16x4_f32(
          false, a, false, b, (short)0, acc[t], false, false);
    }
  }

  int mrow = (m_tile << 4) + ((lane >> 4) << 3);
#pragma unroll
  for (int t = 0; t < NT; ++t) {
    int n = t * 16 + nloc;
    if (n < NCOLS) {
#pragma unroll
      for (int r = 0; r < 8; ++r)
        out[(size_t)(mrow + r) * NCOLS + n] = acc[t][r];
    }
  }
}

// ---------------------------------------------------------------------------
// Depthwise 3x3, SAME, zero-pad. Input x is channel-last (B*P, 64).
// Output dwT channel-last (B*P, 64) = dw.transpose(0,2,3,1) flattened.
// ---------------------------------------------------------------------------
__global__ __launch_bounds__(256) void dwconv_k(
    const float* __restrict__ x, const float* __restrict__ w,
    const float* __restrict__ bias, float* __restrict__ out)
{
  int c  = threadIdx.x & 63;
  int pr = threadIdx.x >> 6;                 // 4 pixels per block
  int bp = blockIdx.x * 4 + pr;              // b*P + p
  int b  = bp >> 14;
  int p  = bp & (PP - 1);
  int h  = p >> 7, wx = p & 127;

  float acc = bias[c];
  const float* wr = w + c * 9;               // dw_w[c,0,:,:]
  const float* xb = x + (size_t)b * PP * 64;
#pragma unroll
  for (int dy = -1; dy <= 1; ++dy) {
    int hh = h + dy;
    if (hh < 0 || hh >= HH) continue;
#pragma unroll
    for (int dx = -1; dx <= 1; ++dx) {
      int wwx = wx + dx;
      if (wwx < 0 || wwx >= WW) continue;
      acc += xb[(size_t)(hh * WW + wwx) * 64 + c] * wr[(dy + 1) * 3 + (dx + 1)];
    }
  }
  out[(size_t)bp * 64 + c] = acc;
}

// ---------------------------------------------------------------------------
// DCN bilinear gather core. val: (B*P, 64) with col = g*32+d.
// om: (B*P, 56): per group g, cols g*27+[0..17] = offsets (K,2),
// g*27+18+[0..8] = mask. Output core: (B*P, 64), col = g*32+d.
// One wave per (b,p,g); lanes span D=32 -> each tap load = one 128B line.
// ---------------------------------------------------------------------------
__global__ __launch_bounds__(256) void dcn_core_k(
    const float* __restrict__ val, const float* __restrict__ om,
    float* __restrict__ core)
{
  int lane = threadIdx.x & 31;   // d
  int wv   = threadIdx.x >> 5;   // 0..7
  int g    = wv & 1;
  int psub = wv >> 1;            // 0..3
  int bp   = blockIdx.x * 4 + psub;
  int b    = bp >> 14;
  int p    = bp & (PP - 1);
  int h    = p >> 7, wx = p & 127;

  const float* omr = om + (size_t)bp * 56 + g * 27;
  const float* vb  = val + (size_t)b * PP * 64 + g * 32 + lane;

  float acc = 0.0f;
#pragma unroll
  for (int kk = 0; kk < 9; ++kk) {
    float ox = omr[2 * kk];
    float oy = omr[2 * kk + 1];
    float mk = omr[18 + kk];
    float ixf = (float)wx + ((float)(kk % 3 - 1) + ox) * 2.0f;
    float iyf = (float)h  + ((float)(kk / 3 - 1) + oy) * 2.0f;
    float x0 = floorf(ixf), y0 = floorf(iyf);
    float tx = ixf - x0,    ty = iyf - y0;
#pragma unroll
    for (int corner = 0; corner < 4; ++corner) {
      int dyc = corner >> 1, dxc = corner & 1;
      float wgt = (dyc ? ty : 1.0f - ty) * (dxc ? tx : 1.0f - tx);
      float xc = x0 + (float)dxc, yc = y0 + (float)dyc;
      float validf = (xc >= 0.0f && xc < (float)WW &&
                      yc >= 0.0f && yc < (float)HH) ? 1.0f : 0.0f;
      int xi = (int)fminf(fmaxf(xc, 0.0f), (float)(WW - 1));
      int yi = (int)fminf(fmaxf(yc, 0.0f), (float)(HH - 1));
      float v = vb[(size_t)(yi * WW + xi) * 64];
      acc += wgt * mk * validf * v;
    }
  }
  core[(size_t)bp * 64 + g * 32 + lane] = acc;
}

// ---------------------------------------------------------------------------
// Epilogue: buf is (B,P,64) flat == (B,64,128,128) reinterpret.
// z = 0.5*(ch[0:32] + ch[32:64]) -> BN -> ReLU. Output (B,32,128,128).
// ---------------------------------------------------------------------------
__global__ __launch_bounds__(256) void finalize_k(
    const float* __restrict__ buf,
    const float* __restrict__ bn_g, const float* __restrict__ bn_b,
    const float* __restrict__ bn_m, const float* __restrict__ bn_v,
    float* __restrict__ out)
{
  int i   = blockIdx.x * 256 + threadIdx.x;     // over B*32*P = 4194304
  int b   = i >> 19;                            // / (32*16384)
  int rem = i & ((1 << 19) - 1);
  int c2  = rem >> 14;
  int q   = rem & 16383;
  size_t base = ((size_t)b << 20) + ((size_t)c2 << 14) + (size_t)q;
  float z = 0.5f * (buf[base] + buf[base + (size_t)(32 << 14)]);
  z = bn_g[c2] * (z - bn_m[c2]) * rsqrtf(bn_v[c2] + 1e-5f) + bn_b[c2];
  out[i] = fmaxf(z, 0.0f);
}

// ---------------------------------------------------------------------------
extern "C" void kernel_launch(void* const* d_in, const int* in_sizes, int n_in,
                              void* d_out, int out_size, void* d_ws, size_t ws_size,
                              hipStream_t stream) {
  (void)in_sizes; (void)n_in; (void)out_size; (void)ws_size;
  const float* x    = (const float*)d_in[0];
  const float* vp_w = (const float*)d_in[1];
  const float* vp_b = (const float*)d_in[2];
  const float* dw_w = (const float*)d_in[3];
  const float* dw_b = (const float*)d_in[4];
  const float* om_w = (const float*)d_in[5];
  const float* om_b = (const float*)d_in[6];
  const float* op_w = (const float*)d_in[7];
  const float* op_b = (const float*)d_in[8];
  const float* bn_g = (const float*)d_in[9];
  const float* bn_b = (const float*)d_in[10];
  const float* bn_m = (const float*)d_in[11];
  const float* bn_v = (const float*)d_in[12];

  float* ws   = (float*)d_ws;
  float* val  = ws;                 // 8,388,608 floats (32 MB)
  float* dwT  = ws + 8388608;       // 8,388,608 floats (32 MB)
  float* om   = ws + 16777216;      // 7,340,032 floats (28 MB)
  float* core = dwT;                // reuse: dwT dead after om GEMM
  float* buf  = val;                // reuse: val dead after DCN core
  float* out  = (float*)d_out;

  const int M = 8 * PP;             // 131072 rows; 8192 M-tiles / 8 waves

  // 1) value projection: val = x(B*P,64) @ vp_w^T + vp_b
  gemm_nt_bias<64><<<1024, 256, 0, stream>>>(x, vp_w, vp_b, val, M);
  // 2) depthwise 3x3 (channel-last in/out)
  dwconv_k<<<32768, 256, 0, stream>>>(x, dw_w, dw_b, dwT);
  // 3) offset/mask projection: om = dwT @ om_w^T + om_b  (56 cols)
  gemm_nt_bias<56><<<1024, 256, 0, stream>>>(dwT, om_w, om_b, om, M);
  // 4) deformable bilinear gather core
  dcn_core_k<<<32768, 256, 0, stream>>>(val, om, core);
  // 5) output projection: buf = core @ op_w^T + op_b
  gemm_nt_bias<64><<<1024, 256, 0, stream>>>(core, op_w, op_b, buf, M);
  // 6) channel-halving + BN + ReLU
  finalize_k<<<16384, 256, 0, stream>>>(buf, bn_g, bn_b, bn_m, bn_v, out);
}